// SUQ_Activation_Full_62345745269514
// MI455X (gfx1250) — compile-verified
//
#include <hip/hip_runtime.h>
#include <hip/hip_bf16.h>
#include <stdint.h>

// SUQ activation moment propagation:
//   a_mean = gelu(h_mean)                                  [B,D]
//   nabla  = gelu'(h_mean)                                  [B,D]  (scratch)
//   a_cov[n,i,l] = nabla[n,i] * h_cov[n,i,l] * nabla[n,l]   [B,D,D]
//
// Memory-bound: ~1.07 GB of traffic -> ~46us floor at 23.3 TB/s.
// Strategy: stage nabla[n,:] (4KB) in LDS via CDNA5 async global->LDS copy,
// then stream h_cov/a_cov with B128 non-temporal loads/stores.

typedef __attribute__((ext_vector_type(4))) float v4f;

#ifndef TPB
#define TPB 256      // 8 waves (wave32)
#endif
#define ROWS 16      // output rows per block

// ---------------------------------------------------------------------------
// Kernel 1: a_mean = gelu(x), nabla = gelu'(x)   (exact erf formulation)
//   gelu(x)  = x * Phi(x),  Phi(x) = 0.5*(1+erf(x/sqrt(2)))
//   gelu'(x) = Phi(x) + x * phi(x),  phi(x) = exp(-x^2/2)/sqrt(2*pi)
// ---------------------------------------------------------------------------
__global__ void __launch_bounds__(TPB)
gelu_mean_nabla_kernel(const float* __restrict__ h_mean,
                       float* __restrict__ a_mean,
                       float* __restrict__ nabla,
                       int n)
{
    int idx = blockIdx.x * TPB + threadIdx.x;
    if (idx < n) {
        float x   = h_mean[idx];
        float Phi = 0.5f * (1.0f + erff(x * 0.70710678118654752440f));
        float phi = 0.39894228040143267794f * expf(-0.5f * x * x);
        a_mean[idx] = x * Phi;
        nabla[idx]  = Phi + x * phi;  // fma
    }
}

// ---------------------------------------------------------------------------
// Kernel 2: a_cov[n,i,l] = nabla[n,i] * h_cov[n,i,l] * nabla[n,l]
//
// Grid: B * (D/ROWS) blocks; block owns rows [i0, i0+ROWS) of batch n.
// nabla[n,0:D] is copied to LDS with global_load_async_to_lds_b128 (ASYNCcnt),
// then each thread streams a fixed 4-column slice across ROWS rows.
// Requires D % (4*TPB) == 0 and D % ROWS == 0 (true for D=1024).
// ---------------------------------------------------------------------------
__global__ void __launch_bounds__(TPB)
cov_sandwich_kernel(const float* __restrict__ h_cov,
                    const float* __restrict__ nabla,
                    float* __restrict__ a_cov,
                    int D)
{
    extern __shared__ float s_nabla[];   // D floats (4KB for D=1024)

    const int tiles_per_batch = D / ROWS;
    const int n   = blockIdx.x / tiles_per_batch;
    const int i0  = (blockIdx.x % tiles_per_batch) * ROWS;
    const int tid = threadIdx.x;

    // --- Async copy nabla[n, :] -> LDS (CDNA5 async-tensor data path). ---
    // Each lane issues one B128 (16B) transfer per 4*TPB columns.
    for (int c = 4 * tid; c < D; c += 4 * TPB) {
        const float* gsrc = nabla + (size_t)n * (size_t)D + c;
        // Low 32 bits of a flat LDS-aperture address ARE the LDS byte offset.
        uint32_t lds_off = (uint32_t)(uintptr_t)(&s_nabla[c]);
        asm volatile("global_load_async_to_lds_b128 %0, %1, off"
                     :: "v"(lds_off), "v"(gsrc)
                     : "memory");
    }
    asm volatile("s_wait_asynccnt 0x0" ::: "memory");  // my wave's copies done
    __syncthreads();                                    // all waves' copies visible

    // --- Stream ROWS rows of the 16-byte column slice owned by this thread. ---
    for (int c = 4 * tid; c < D; c += 4 * TPB) {
        // Column scales: one conflict-free LDS B128 load, reused for all rows.
        v4f cscale = *(const v4f*)(&s_nabla[c]);

        const size_t rowbase = ((size_t)n * (size_t)D + (size_t)i0) * (size_t)D
                             + (size_t)c;
        const float* src = h_cov + rowbase;
        float*       dst = a_cov + rowbase;

#pragma unroll
        for (int r = 0; r < ROWS; ++r) {
            // Row scale: LDS broadcast (all lanes same address).
            float si = s_nabla[i0 + r];
            // Streamed exactly once each -> non-temporal to spare the 192MB L2.
            v4f h = __builtin_nontemporal_load(
                        (const v4f*)(src + (size_t)r * (size_t)D));
            v4f o = (cscale * h) * si;
            __builtin_nontemporal_store(
                o, (v4f*)(dst + (size_t)r * (size_t)D));
        }
    }
}

// ---------------------------------------------------------------------------
// Host-side launch
// d_in[0] = h_mean [B,D] f32, d_in[1] = h_cov [B,D,D] f32
// d_out   = [a_mean (B*D) | a_cov (B*D*D)] f32
// d_ws    = nabla (B*D floats)
// ---------------------------------------------------------------------------
extern "C" void kernel_launch(void* const* d_in, const int* in_sizes, int n_in,
                              void* d_out, int out_size, void* d_ws, size_t ws_size,
                              hipStream_t stream)
{
    const float* h_mean = (const float*)d_in[0];
    const float* h_cov  = (const float*)d_in[1];

    const int BD  = in_sizes[0];                 // B*D      = 131072
    const int D   = in_sizes[1] / in_sizes[0];   // 1024
    const int B   = BD / D;                      // 128

    float* a_mean = (float*)d_out;
    float* a_cov  = (float*)d_out + BD;
    float* nabla  = (float*)d_ws;

    // Kernel 1: gelu mean + derivative (tiny).
    {
        int blocks = (BD + TPB - 1) / TPB;
        gelu_mean_nabla_kernel<<<blocks, TPB, 0, stream>>>(h_mean, a_mean, nabla, BD);
    }

    // Kernel 2: covariance sandwich (the 1 GB stream).
    {
        int blocks = B * (D / ROWS);             // 8192 blocks
        size_t lds_bytes = (size_t)D * sizeof(float);
        cov_sandwich_kernel<<<blocks, TPB, lds_bytes, stream>>>(h_cov, nabla, a_cov, D);
    }
}